// ARMANet_10737418240589
// MI455X (gfx1250) — compile-verified
//
#include <hip/hip_runtime.h>
#include <hip/hip_bf16.h>
#include <stdint.h>

typedef float v2f __attribute__((ext_vector_type(2)));
typedef float v8f __attribute__((ext_vector_type(8)));

#define HID 64
#define OUTD 32
#define NGRAPH 64

// ---------------------------------------------------------------------------
// generic zero
__global__ void k_zero(float* p, size_t n) {
    size_t i = (size_t)blockIdx.x * blockDim.x + threadIdx.x;
    if (i < n) p[i] = 0.0f;
}

// ---------------------------------------------------------------------------
// degree of target nodes: deg[col[e]] += 1
__global__ void k_degree(const long long* ei, float* deg, int E) {
    int e = blockIdx.x * blockDim.x + threadIdx.x;
    if (e < E) {
        int col = (int)ei[(size_t)E + e];
        atomicAdd(&deg[col], 1.0f);
    }
}

__global__ void k_dinv(const float* deg, float* dinv, int N) {
    int n = blockIdx.x * blockDim.x + threadIdx.x;
    if (n < N) {
        float d = deg[n];
        dinv[n] = (d > 0.0f) ? rsqrtf(fmaxf(d, 1.0f)) : 0.0f;
    }
}

// ---------------------------------------------------------------------------
// C[N,64] = A[N,64] @ W[64,64]   (fp32 WMMA 16x16x4, wave32)
// one wave -> 16 rows x 64 cols; 8 waves / block -> 128 rows / block
__global__ void k_gemm64(const float* __restrict__ A,
                         const float* __restrict__ W,
                         float* __restrict__ C, int N) {
    __shared__ float sW[HID * HID];   // 16 KB
    const int tid  = threadIdx.x;
    for (int i = tid; i < HID * HID; i += 256) sW[i] = W[i];
    __syncthreads();

    const int wave    = tid >> 5;
    const int lane    = tid & 31;
    const int half    = lane >> 4;           // 0: lanes 0-15, 1: lanes 16-31
    const int l       = lane & 15;
    const int rowBase = (blockIdx.x * 8 + wave) * 16;
    if (rowBase + 16 > N) return;            // wave-uniform; N % 16 == 0 here

    v8f acc[4];
    acc[0] = {}; acc[1] = {}; acc[2] = {}; acc[3] = {};

    const float* arow = A + (size_t)(rowBase + l) * HID;

    #pragma unroll
    for (int k0 = 0; k0 < HID; k0 += 4) {
        // A fragment: VGPR0 = K=k0 (lo lanes) / k0+2 (hi lanes); VGPR1 = +1
        v2f a;
        a.x = arow[k0 + 2 * half];
        a.y = arow[k0 + 2 * half + 1];
        #pragma unroll
        for (int nt = 0; nt < 4; ++nt) {
            const int n0 = nt * 16;
            // B fragment: VGPR0 = row k0 (lo) / k0+2 (hi); VGPR1 = +1
            v2f b;
            b.x = sW[(k0 + 2 * half) * HID + n0 + l];
            b.y = sW[(k0 + 2 * half + 1) * HID + n0 + l];
            acc[nt] = __builtin_amdgcn_wmma_f32_16x16x4_f32(
                false, a, false, b, (short)0, acc[nt], false, false);
        }
    }

    // C/D layout: VGPR r -> row rowBase+r (+8 for hi lanes), col n0+l
    #pragma unroll
    for (int nt = 0; nt < 4; ++nt) {
        const int n0  = nt * 16;
        const int row = rowBase + 8 * half;
        #pragma unroll
        for (int r = 0; r < 8; ++r)
            C[(size_t)(row + r) * HID + n0 + l] = acc[nt][r];
    }
}

// ---------------------------------------------------------------------------
// scatter: dst[col] += dinv[row]*dinv[col] * src[row]   (16 lanes/edge, float4)
__global__ void k_scatter(const long long* __restrict__ ei,
                          const float* __restrict__ dinv,
                          const float* __restrict__ src,
                          float* __restrict__ dst, int E) {
    size_t t = (size_t)blockIdx.x * blockDim.x + threadIdx.x;
    size_t total = (size_t)E * 16;
    if (t >= total) return;
    int e = (int)(t >> 4);
    int q = (int)(t & 15);          // feature chunk: 4 floats
    int row = (int)ei[e];
    int col = (int)ei[(size_t)E + e];
    float norm = dinv[row] * dinv[col];
    if (norm == 0.0f) return;
    const float4 v = *(const float4*)(src + (size_t)row * HID + q * 4);
    float* d = dst + (size_t)col * HID + q * 4;
    atomicAdd(d + 0, v.x * norm);
    atomicAdd(d + 1, v.y * norm);
    atomicAdd(d + 2, v.z * norm);
    atomicAdd(d + 3, v.w * norm);
}

// ---------------------------------------------------------------------------
// h = relu(agg + root + b)
__global__ void k_combine(const float* __restrict__ agg,
                          const float* __restrict__ root,
                          const float* __restrict__ b,
                          float* __restrict__ h, size_t n) {
    size_t i = (size_t)blockIdx.x * blockDim.x + threadIdx.x;
    if (i < n) {
        float v = agg[i] + root[i] + b[i & (HID - 1)];
        h[i] = v > 0.0f ? v : 0.0f;
    }
}

// ---------------------------------------------------------------------------
// global mean pool (sums + counts); batch is sorted so a 128-node block
// touches few graphs -> LDS-accumulate then sparse global atomics
__global__ void k_pool(const float* __restrict__ h,
                       const long long* __restrict__ batch,
                       float* __restrict__ pooled, float* __restrict__ cnt,
                       int N) {
    __shared__ float sP[NGRAPH * HID];   // 16 KB
    __shared__ float sC[NGRAPH];
    const int tid = threadIdx.x;
    for (int i = tid; i < NGRAPH * HID; i += 256) sP[i] = 0.0f;
    if (tid < NGRAPH) sC[tid] = 0.0f;
    __syncthreads();

    const int base = blockIdx.x * 128;
    if (tid < 128) {
        int n = base + tid;
        if (n < N) atomicAdd(&sC[(int)batch[n]], 1.0f);
    }
    #pragma unroll 4
    for (int i = 0; i < 32; ++i) {         // 128 nodes * 64 feats / 256 thr
        int e = i * 256 + tid;
        int n = base + (e >> 6);
        if (n < N) {
            int g = (int)batch[n];
            atomicAdd(&sP[g * HID + (e & 63)], h[(size_t)n * HID + (e & 63)]);
        }
    }
    __syncthreads();
    for (int i = tid; i < NGRAPH * HID; i += 256) {
        float v = sP[i];
        if (v != 0.0f) atomicAdd(&pooled[i], v);
    }
    if (tid < NGRAPH) {
        float v = sC[tid];
        if (v != 0.0f) atomicAdd(&cnt[tid], v);
    }
}

// ---------------------------------------------------------------------------
// out[g,o] = (pooled_sum[g,:]/cnt[g]) @ fc_w + fc_b   (64 graphs x 32 outs)
__global__ void k_fc(const float* __restrict__ pooled,
                     const float* __restrict__ cnt,
                     const float* __restrict__ fcw,
                     const float* __restrict__ fcb,
                     float* __restrict__ out) {
    int g = blockIdx.x;        // 64
    int o = threadIdx.x;       // 32
    float inv = 1.0f / fmaxf(cnt[g], 1.0f);
    float s = fcb[o];
    #pragma unroll
    for (int c = 0; c < HID; ++c)
        s += pooled[g * HID + c] * inv * fcw[c * OUTD + o];
    out[g * OUTD + o] = s;
}

// ---------------------------------------------------------------------------
extern "C" void kernel_launch(void* const* d_in, const int* in_sizes, int n_in,
                              void* d_out, int out_size, void* d_ws, size_t ws_size,
                              hipStream_t stream) {
    const float*     x    = (const float*)d_in[0];
    const long long* ei   = (const long long*)d_in[1];
    const long long* bat  = (const long long*)d_in[2];
    const float*     w1i  = (const float*)d_in[3];
    const float*     w1r  = (const float*)d_in[4];
    const float*     b1   = (const float*)d_in[5];
    const float*     w2i  = (const float*)d_in[6];
    const float*     w2r  = (const float*)d_in[7];
    const float*     b2   = (const float*)d_in[8];
    const float*     fcw  = (const float*)d_in[9];
    const float*     fcb  = (const float*)d_in[10];
    float*           out  = (float*)d_out;

    const int N = in_sizes[0] / HID;
    const int E = in_sizes[1] / 2;
    const size_t NN  = (size_t)N;
    const size_t NF  = NN * HID;

    float* ws     = (float*)d_ws;
    float* deg    = ws;
    float* dinv   = ws + NN;
    float* bufA   = ws + 2 * NN;             // gemm-init out / h2
    float* bufR   = bufA + NF;               // gemm-root out
    float* bufG   = bufR + NF;               // aggregation
    float* bufH   = bufG + NF;               // h after layer 1
    float* pooled = bufH + NF;
    float* cnt    = pooled + NGRAPH * HID;

    const int ZB = 256;
    auto zgrid = [](size_t n) { return (unsigned)((n + 255) / 256); };

    const int rowTiles  = N / 16;                       // N % 16 == 0 (100000)
    const int gemmGrid  = (rowTiles + 7) / 8;
    const int scatGrid  = (int)(((size_t)E * 16 + 255) / 256);
    const int poolGrid  = (N + 127) / 128;

    // degree + dinv
    k_zero<<<zgrid(NN), ZB, 0, stream>>>(deg, NN);
    k_degree<<<(E + 255) / 256, ZB, 0, stream>>>(ei, deg, E);
    k_dinv<<<zgrid(NN), ZB, 0, stream>>>(deg, dinv, N);

    // ---- layer 1 ----
    k_gemm64<<<gemmGrid, 256, 0, stream>>>(x, w1i, bufA, N);
    k_gemm64<<<gemmGrid, 256, 0, stream>>>(x, w1r, bufR, N);
    k_zero<<<zgrid(NF), ZB, 0, stream>>>(bufG, NF);
    k_scatter<<<scatGrid, 256, 0, stream>>>(ei, dinv, bufA, bufG, E);
    k_combine<<<zgrid(NF), ZB, 0, stream>>>(bufG, bufR, b1, bufH, NF);

    // ---- layer 2 ----
    k_gemm64<<<gemmGrid, 256, 0, stream>>>(bufH, w2i, bufA, N);
    k_gemm64<<<gemmGrid, 256, 0, stream>>>(bufH, w2r, bufR, N);
    k_zero<<<zgrid(NF), ZB, 0, stream>>>(bufG, NF);
    k_scatter<<<scatGrid, 256, 0, stream>>>(ei, dinv, bufA, bufG, E);
    k_combine<<<zgrid(NF), ZB, 0, stream>>>(bufG, bufR, b2, bufA, NF);

    // ---- pool + fc ----
    k_zero<<<zgrid(NGRAPH * HID + NGRAPH), ZB, 0, stream>>>(pooled, NGRAPH * HID + NGRAPH);
    k_pool<<<poolGrid, 256, 0, stream>>>(bufA, bat, pooled, cnt, N);
    k_fc<<<NGRAPH, OUTD, 0, stream>>>(pooled, cnt, fcw, fcb, out);
}